// CapsuleSubLayer_24704651887087
// MI455X (gfx1250) — compile-verified
//
#include <hip/hip_runtime.h>

// CapsNet routing sublayer, algebraically collapsed:
//   xs[c,i]      = sum_b x[b,i,c]
//   u_sum[c,u,s] = sum_i W[c,u,s,i] * xs[c,i]        (WMMA f32 16x16x4 matvec)
//   3x routing on u_sum only (u_hat never materialized; it is provably unused)
// Output: v[c,u,s]  (C=1152, U=10, S=16)

typedef __attribute__((ext_vector_type(2))) float v2f;
typedef __attribute__((ext_vector_type(8))) float v8f;

#define B_SZ   256
#define I_SZ   8
#define C_SZ   1152
#define U_SZ   10
#define S_SZ   16
#define IC     (I_SZ * C_SZ)      // 9216
#define NCHUNK 16
#define CHUNK  (B_SZ / NCHUNK)    // 16

// ---------- Kernel A1: partial batch reduction (deterministic, no atomics) ----------
__global__ void caps_batch_partial(const float* __restrict__ x, float* __restrict__ partial) {
    const int p = blockIdx.x * blockDim.x + threadIdx.x;   // p = i*1152 + c, coalesced in c
    const int k = blockIdx.y;                              // batch chunk
    const float* xp = x + (size_t)k * CHUNK * IC + p;
    float s = 0.0f;
#pragma unroll
    for (int b = 0; b < CHUNK; ++b) s += xp[(size_t)b * IC];
    partial[k * IC + p] = s;
}

// ---------- Kernel A2: finalize xs[i*1152 + c] ----------
__global__ void caps_batch_final(const float* __restrict__ partial, float* __restrict__ xs) {
    const int p = blockIdx.x * blockDim.x + threadIdx.x;
    float s = 0.0f;
#pragma unroll
    for (int k = 0; k < NCHUNK; ++k) s += partial[k * IC + p];
    xs[p] = s;
}

// ---------- Kernel B: one wave32 per channel c: WMMA matvec + dynamic routing ----------
__global__ void __launch_bounds__(32) caps_route(const float* __restrict__ weight,
                                                 const float* __restrict__ xs_ic,
                                                 float* __restrict__ out) {
    const int c  = blockIdx.x;        // 0..1151
    const int l  = threadIdx.x;       // 0..31 (one wave)
    const int hi = l >> 4;            // lane half: selects K pair / M offset
    const int m  = l & 15;            // row within 16-row tile

    __shared__ float usum[U_SZ * S_SZ];   // 160 floats
    __shared__ float bl[16];              // routing logits b[u], 10 used

    // xs for this channel (uniform per block): xs_ic[i*1152 + c]
    float xsv[I_SZ];
#pragma unroll
    for (int i = 0; i < I_SZ; ++i) xsv[i] = xs_ic[i * C_SZ + c];

    // B operand (4x16, columns broadcast): half0 lanes carry K={kb,kb+1}, half1 K={kb..}
    const int kb = hi ? 2 : 0;
    v2f bv0, bv1;
    bv0.x = xsv[kb];     bv0.y = xsv[kb + 1];       // K-step 0 (k = 0..3)
    bv1.x = xsv[kb + 4]; bv1.y = xsv[kb + 5];       // K-step 1 (k = 4..7)

    const float* Wc = weight + (size_t)c * (U_SZ * S_SZ * I_SZ);

    // Tile t == capsule u == rows (u, s=0..15); K=8 split into two 16x16x4 WMMAs.
#pragma unroll
    for (int t = 0; t < U_SZ; ++t) {
        const float* Wr = Wc + (t * 16 + m) * I_SZ;     // row (u=t, s=m), 8 contiguous floats
        v2f a0, a1;
        a0.x = Wr[kb];     a0.y = Wr[kb + 1];
        a1.x = Wr[kb + 4]; a1.y = Wr[kb + 5];
        v8f acc = {};
        acc = __builtin_amdgcn_wmma_f32_16x16x4_f32(false, a0, false, bv0,
                                                    (short)0, acc, false, false);
        acc = __builtin_amdgcn_wmma_f32_16x16x4_f32(false, a1, false, bv1,
                                                    (short)0, acc, false, false);
        // D: vgpr r -> M=r (lanes 0-15) / M=r+8 (lanes 16-31); columns are replicas.
        if (m == 0) {
#pragma unroll
            for (int j = 0; j < 8; ++j) usum[t * 16 + hi * 8 + j] = acc[j];
        }
    }
    if (l < U_SZ) bl[l] = 0.0f;
    __syncthreads();

    // Routing layout: element e = l + 32r  ->  u = 2r + hi, s = l&15.
    // Each u occupies one 16-lane half at one register index -> shfl_xor{1,2,4,8} reductions.
    float us[5], vv[5];
#pragma unroll
    for (int r = 0; r < 5; ++r) us[r] = usum[l + 32 * r];

    for (int it = 0; it < 3; ++it) {
        // softmax over U=10 (each lane redundantly, reads LDS logits)
        float bloc[U_SZ];
        float bmax = -3.402823466e+38f;
#pragma unroll
        for (int u = 0; u < U_SZ; ++u) { bloc[u] = bl[u]; bmax = fmaxf(bmax, bloc[u]); }
        float den = 0.0f;
#pragma unroll
        for (int u = 0; u < U_SZ; ++u) { bloc[u] = expf(bloc[u] - bmax); den += bloc[u]; }
        const float rden = 1.0f / den;

#pragma unroll
        for (int r = 0; r < 5; ++r) {
            const int u = 2 * r + hi;
            const float cij = bloc[u] * rden;
            const float sj  = cij * us[r];
            // mag_sq = sum_s sj^2 over this u's 16 lanes
            float t = sj * sj;
            t += __shfl_xor(t, 1, 32);
            t += __shfl_xor(t, 2, 32);
            t += __shfl_xor(t, 4, 32);
            t += __shfl_xor(t, 8, 32);
            const float mag = sqrtf(t);
            const float v   = (t / (1.0f + t)) * (sj / mag);   // squash
            vv[r] = v;
            // u_vj1[u] = sum_s u_sum * v / B   (batch sum already folded into u_sum)
            float d = us[r] * v;
            d += __shfl_xor(d, 1, 32);
            d += __shfl_xor(d, 2, 32);
            d += __shfl_xor(d, 4, 32);
            d += __shfl_xor(d, 8, 32);
            if (m == 0) bl[u] += d * (1.0f / (float)B_SZ);   // lanes 0 & 16, disjoint u
        }
        __syncthreads();
    }

#pragma unroll
    for (int r = 0; r < 5; ++r) out[(size_t)c * (U_SZ * S_SZ) + l + 32 * r] = vv[r];
}

extern "C" void kernel_launch(void* const* d_in, const int* in_sizes, int n_in,
                              void* d_out, int out_size, void* d_ws, size_t ws_size,
                              hipStream_t stream) {
    const float* x      = (const float*)d_in[0];   // [256, 8, 1152]
    const float* weight = (const float*)d_in[1];   // [1, 1152, 10, 16, 8]
    float*       out    = (float*)d_out;           // [1152, 10, 16]

    float* wsf     = (float*)d_ws;
    float* partial = wsf;                 // NCHUNK * 9216 floats
    float* xs      = wsf + NCHUNK * IC;   // 9216 floats

    dim3 gA1(IC / 256, NCHUNK);
    caps_batch_partial<<<gA1, 256, 0, stream>>>(x, partial);
    caps_batch_final<<<IC / 256, 256, 0, stream>>>(partial, xs);
    caps_route<<<C_SZ, 32, 0, stream>>>(weight, xs, out);
}